// HyperbolicGraphConv_13194139533843
// MI455X (gfx1250) — compile-verified
//
#include <hip/hip_runtime.h>
#include <math.h>

// Hyperbolic graph conv for MI455X (gfx1250):
//   K1: expmap   — row norms via V_WMMA_F32_16X16X4_F32 Gram diagonal, scale rows
//   K2: zero     — clear accumulator (d_out)
//   K3: scatter  — per-edge gather (L2-resident x_proj) + native f32 atomics,
//                  NT hints on the once-streamed edge lists
//   K4: logmap   — Gram-diagonal norms again, atanh scale + bias, in place
//
// fp32 throughout: the op is L2-bandwidth/atomic bound (~0.25 GFLOP vs ~70 MB
// HBM + ~410 MB L2 gather traffic), and tanh/atanh of the norm is precision
// sensitive, so low-precision WMMA would not help and would hurt accuracy.
// Tile staging now uses the CDNA5 async global->LDS path (ASYNCcnt) instead of
// the global->VGPR->ds_store round trip.

typedef __attribute__((ext_vector_type(2))) float v2f;
typedef __attribute__((ext_vector_type(4))) float v4f;
typedef __attribute__((ext_vector_type(8))) float v8f;

#define NWAVES 8          // waves per 256-thread block (wave32)
#define TSTR   68         // LDS tile row stride in floats (64 + 4 pad: no bank conflicts)
#define EPSF   1e-8f

// One wave processes one 16-row x 64-col tile:
//   - async-stage tile to LDS (global_load_async_to_lds_b128, ASYNCcnt)
//   - Gram = X*X^T via 16 chained V_WMMA_F32_16X16X4_F32 (A == B registers:
//     the 16x4 A layout and 4x16 B layout are lane-transposes of each other)
//   - diagonal -> squared row norms -> tanh/atanh scale
//   - scaled (+bias) rows written back with coalesced 128-bit stores
template <bool IS_EXP>
__global__ __launch_bounds__(256) void norm_scale_kernel(
    const float* __restrict__ in, float* __restrict__ out,
    const float* __restrict__ bias, int N)
{
  __shared__ float s_tile[NWAVES][16 * TSTR];
  __shared__ float s_diag[NWAVES][32 * 8];
  __shared__ float s_scl [NWAVES][16];

  const int lane = threadIdx.x & 31;
  const int wv   = threadIdx.x >> 5;
  const int unit = blockIdx.x * NWAVES + wv;
  const int ntiles = N >> 4;
  const int rem    = N & 15;

  float* tile = s_tile[wv];
  float* diag = s_diag[wv];
  float* scl  = s_scl[wv];

  if (unit < ntiles) {                      // wave-uniform branch: EXEC all 1s
    const int node0 = unit << 4;
    const float* gsrc = in + (size_t)node0 * 64;

    // Async-stage the 16x64 tile into padded LDS: per-lane 16B directly to
    // LDS (no VGPR round trip), 512B per issue across the wave.
#pragma unroll
    for (int i = 0; i < 8; ++i) {
      const int idx = i * 128 + lane * 4;   // element index inside tile
      const int row = idx >> 6, col = idx & 63;
      const unsigned ldsb =
          (unsigned)(size_t)(const void*)&tile[row * TSTR + col];
      const unsigned long long ga =
          (unsigned long long)(size_t)(const void*)(gsrc + idx);
      asm volatile("global_load_async_to_lds_b128 %0, %1, off"
                   :: "v"(ldsb), "v"(ga) : "memory");
    }
    asm volatile("s_wait_asynccnt 0x0" ::: "memory");

    // Gram accumulation on the matrix pipe.
    v8f acc = {0.f, 0.f, 0.f, 0.f, 0.f, 0.f, 0.f, 0.f};
    const int arow = lane & 15;             // A: lane holds row (lane&15)
    const int koff = (lane >> 4) << 1;      //    upper half-wave holds K+2,K+3
#pragma unroll
    for (int k0 = 0; k0 < 64; k0 += 4) {
      v2f a = *(const v2f*)&tile[arow * TSTR + k0 + koff];
      acc = __builtin_amdgcn_wmma_f32_16x16x4_f32(
          /*neg_a=*/false, a, /*neg_b=*/false, a,
          /*c_mod=*/(short)0, acc, /*reuse_a=*/false, /*reuse_b=*/false);
    }

    // Spill accumulator to LDS, read back the Gram diagonal.
#pragma unroll
    for (int r = 0; r < 8; ++r) diag[lane * 8 + r] = acc[r];

    if (lane < 16) {
      const int comp = (lane < 8) ? lane : lane - 8;   // D[m][m]: VGPR m%8,
      const int sl   = (lane < 8) ? lane : lane + 8;   // lane m (m<8) else m+8
      const float nsq = diag[sl * 8 + comp];
      const float n = sqrtf(fmaxf(nsq, 0.f));
      float s;
      if (IS_EXP) s = tanhf(n)  / (n + EPSF);
      else        s = atanhf(n) / (n + EPSF);
      scl[lane] = s;
    }

    v4f b4 = {0.f, 0.f, 0.f, 0.f};
    if (!IS_EXP) b4 = *(const v4f*)(bias + (lane & 15) * 4);

    float* gdst = out + (size_t)node0 * 64;
#pragma unroll
    for (int i = 0; i < 8; ++i) {
      const int idx = i * 128 + lane * 4;
      const int row = idx >> 6, col = idx & 63;
      const float s = scl[row];
      v4f v = *(const v4f*)&tile[row * TSTR + col];
      v4f r = v * s;
      if (!IS_EXP) r = r + b4;
      *(v4f*)(gdst + idx) = r;
    }
  } else if (unit == ntiles && rem != 0) {
    // Scalar tail for N % 16 rows (no WMMA here, divergence is fine).
    if (lane < rem) {
      const int rrow = (ntiles << 4) + lane;
      const float* p = in + (size_t)rrow * 64;
      float nsq = 0.f;
      for (int k = 0; k < 64; ++k) { float t = p[k]; nsq += t * t; }
      const float n = sqrtf(nsq);
      const float s = IS_EXP ? (tanhf(n) / (n + EPSF))
                             : (atanhf(n) / (n + EPSF));
      float* q = out + (size_t)rrow * 64;
      for (int k = 0; k < 64; ++k)
        q[k] = p[k] * s + (IS_EXP ? 0.f : bias[k]);
    }
  }
}

__global__ __launch_bounds__(256) void zero_kernel(float* __restrict__ out, int n4)
{
  int i = blockIdx.x * blockDim.x + threadIdx.x;
  const int stride = gridDim.x * blockDim.x;
  const v4f z = {0.f, 0.f, 0.f, 0.f};
  for (; i < n4; i += stride) *(v4f*)(out + (size_t)i * 4) = z;
}

// 16 lanes per edge, one float4 per lane: 256B coalesced gather from
// L2-resident x_proj, then 4 native f32 atomic adds into the accumulator.
// Edge metadata is streamed once -> non-temporal hints keep it from evicting
// the reuse-critical x_proj / accumulator working set out of the 192MB L2.
__global__ __launch_bounds__(256) void scatter_kernel(
    const int* __restrict__ src, const int* __restrict__ dst,
    const float* __restrict__ vals, const float* __restrict__ xp,
    float* __restrict__ out, int E)
{
  const int t = blockIdx.x * 256 + threadIdx.x;
  if (t >= (E << 4)) return;
  const int e = t >> 4;
  const int c = t & 15;
  const int sN = __builtin_nontemporal_load(src + e);
  const int dN = __builtin_nontemporal_load(dst + e);
  const float w = __builtin_nontemporal_load(vals + e);
  v4f v = *(const v4f*)(xp + (size_t)sN * 64 + c * 4);
  float* p = out + (size_t)dN * 64 + c * 4;
  unsafeAtomicAdd(p + 0, v[0] * w);
  unsafeAtomicAdd(p + 1, v[1] * w);
  unsafeAtomicAdd(p + 2, v[2] * w);
  unsafeAtomicAdd(p + 3, v[3] * w);
}

extern "C" void kernel_launch(void* const* d_in, const int* in_sizes, int n_in,
                              void* d_out, int out_size, void* d_ws, size_t ws_size,
                              hipStream_t stream)
{
  (void)n_in; (void)out_size; (void)ws_size;
  const float* x    = (const float*)d_in[0];
  const int*   src  = (const int*)d_in[1];   // jnp.int64 w/o x64 -> int32 on device
  const int*   dst  = (const int*)d_in[2];
  const float* vals = (const float*)d_in[3];
  const float* bias = (const float*)d_in[4];

  const int N = in_sizes[0] / 64;
  const int E = in_sizes[1];

  float* xp  = (float*)d_ws;     // x_proj: N*64 floats (25.6 MB, L2-resident)
  float* out = (float*)d_out;

  const int units   = (N >> 4) + ((N & 15) ? 1 : 0);
  const int nblocks = (units + NWAVES - 1) / NWAVES;

  // K1: expmap into workspace
  norm_scale_kernel<true><<<nblocks, 256, 0, stream>>>(x, xp, nullptr, N);

  // K2: zero the accumulator (d_out is poisoned by the harness)
  const int n4 = N * 16;                      // N*64/4 float4s
  int zb = (n4 + 255) / 256; if (zb > 4096) zb = 4096;
  zero_kernel<<<zb, 256, 0, stream>>>(out, n4);

  // K3: COO scatter-accumulate
  const int total = E * 16;
  scatter_kernel<<<(total + 255) / 256, 256, 0, stream>>>(src, dst, vals, xp, out, E);

  // K4: logmap + bias, in place on d_out
  norm_scale_kernel<false><<<nblocks, 256, 0, stream>>>(out, out, bias, N);
}